// BalancedLoss_77189152244307
// MI455X (gfx1250) — compile-verified
//
#include <hip/hip_runtime.h>
#include <math.h>

#define HW   1024
#define NB   16
#define EPSF 1e-8f

// ---------------------------------------------------------------------------
// CDNA5 async global->LDS path (gfx1250). Guarded so compile always succeeds.
// ---------------------------------------------------------------------------
#if defined(__gfx1250__) && __has_builtin(__builtin_amdgcn_global_load_async_to_lds_b32) && __has_builtin(__builtin_amdgcn_s_wait_asynccnt)
#define USE_ASYNC 1
#else
#define USE_ASYNC 0
#endif

__device__ __forceinline__ void lds_copy_f32(float* dst, const float* src) {
#if USE_ASYNC
  __builtin_amdgcn_global_load_async_to_lds_b32(
      (__attribute__((address_space(1))) int*)src,
      (__attribute__((address_space(3))) int*)dst,
      0, 0);
#else
  *dst = *src;
#endif
}

__device__ __forceinline__ void wait_async_loads() {
#if USE_ASYNC
  __builtin_amdgcn_s_wait_asynccnt(0);
#endif
}

// ws layout: doubles[0..40]:
//   0 Sbce1, 1 Sbce2, 2 S1p, 3 S2p, 4 S1d, 5 S2d, 6 Spd, 7 Scurv,
//   8..23 Sdem[b], 24..39 Sdem2[b], 40 Sh
// uints at byte 41*8: [0] min_gp bits, [1] max_gp, [2] min_gd, [3] max_gd

__global__ void init_ws_k(double* dws, unsigned* uws) {
  int i = threadIdx.x;
  if (i < 41) dws[i] = 0.0;
  if (i == 41) {
    uws[0] = 0x7F800000u; uws[1] = 0u;   // +inf / 0 (g is always > 0)
    uws[2] = 0x7F800000u; uws[3] = 0u;
  }
}

template <int WPAD, int HALO>
__device__ __forceinline__ void load_tile(float* sdst, const float* src,
                                          int x0, int y0) {
  for (int i = threadIdx.x; i < WPAD * WPAD; i += 256) {
    int px = i % WPAD, py = i / WPAD;
    int gx = x0 + px - HALO, gy = y0 + py - HALO;
    if (gx >= 0 && gx < HW && gy >= 0 && gy < HW)
      lds_copy_f32(&sdst[i], &src[(size_t)gy * HW + gx]);
    else
      sdst[i] = 0.0f;   // zero padding of every conv
  }
}

__global__ __launch_bounds__(256)
void pass1_k(const float* __restrict__ pred, const float* __restrict__ target,
             const float* __restrict__ dem, double* dws, unsigned* uws) {
  __shared__ float sT[38 * 38];   // target, halo 3
  __shared__ float sSM[36 * 36];  // box-smoothed, halo 2
  __shared__ float sE[36 * 36];   // edges, halo 2
  __shared__ float sD[34 * 34];   // dilated, halo 1
  __shared__ float sP[34 * 34];   // raw pred, halo 1
  __shared__ float sS[34 * 34];   // sigmoid(pred), halo 1
  __shared__ float sM[34 * 34];   // dem, halo 1
  __shared__ float red[8 * 14];

  const int b = blockIdx.z;
  const int x0 = blockIdx.x * 32, y0 = blockIdx.y * 32;
  const size_t bb = (size_t)b * HW * HW;

  load_tile<38, 3>(sT, target + bb, x0, y0);
  load_tile<34, 1>(sP, pred + bb, x0, y0);
  load_tile<34, 1>(sM, dem + bb, x0, y0);
  wait_async_loads();
  __syncthreads();

  // box smooth (36x36) + sigmoid of pred (34x34, zero for OOB pixels since
  // conv padding applies to pred_prob, not pred)
  for (int i = threadIdx.x; i < 36 * 36; i += 256) {
    int qx = i % 36, qy = i / 36;
    float s = 0.f;
#pragma unroll
    for (int dy = 0; dy < 3; ++dy)
#pragma unroll
      for (int dx = 0; dx < 3; ++dx) s += sT[(qy + dy) * 38 + (qx + dx)];
    sSM[i] = s * (1.0f / 9.0f);
  }
  for (int i = threadIdx.x; i < 34 * 34; i += 256) {
    int px = i % 34, py = i / 34;
    int gx = x0 + px - 1, gy = y0 + py - 1;
    bool inb = (gx >= 0) & (gx < HW) & (gy >= 0) & (gy < HW);
    sS[i] = inb ? (1.0f / (1.0f + expf(-sP[i]))) : 0.0f;
  }
  __syncthreads();

  // edges (36x36), forced 0 at OOB (conv zero-padding of the edges map)
  for (int i = threadIdx.x; i < 36 * 36; i += 256) {
    int qx = i % 36, qy = i / 36;
    int gx = x0 + qx - 2, gy = y0 + qy - 2;
    bool inb = (gx >= 0) & (gx < HW) & (gy >= 0) & (gy < HW);
    float e = (fabsf(sT[(qy + 1) * 38 + (qx + 1)] - sSM[i]) > 0.15f) ? 1.f : 0.f;
    sE[i] = inb ? e : 0.0f;
  }
  __syncthreads();

  // dilated (34x34), forced 0 at OOB
  for (int i = threadIdx.x; i < 34 * 34; i += 256) {
    int qx = i % 34, qy = i / 34;
    int gx = x0 + qx - 1, gy = y0 + qy - 1;
    bool inb = (gx >= 0) & (gx < HW) & (gy >= 0) & (gy < HW);
    float s = 0.f;
#pragma unroll
    for (int dy = 0; dy < 3; ++dy)
#pragma unroll
      for (int dx = 0; dx < 3; ++dx) s += sE[(qy + dy) * 36 + (qx + dx)];
    sD[i] = (inb && s > 0.5f) ? 1.0f : 0.0f;
  }
  __syncthreads();

  // per-pixel fused math, 4 pixels/thread
  float a0 = 0, a1 = 0, a2 = 0, a3 = 0, a4 = 0, a5 = 0, a6 = 0, a7 = 0,
        a8 = 0, a9 = 0;
  float mnp = 3.4e38f, mxp = 0.f, mnd = 3.4e38f, mxd = 0.f;
  const int tx = threadIdx.x & 31;
  const int tyb = threadIdx.x >> 5;
#pragma unroll
  for (int r = 0; r < 4; ++r) {
    int ty = tyb + r * 8;
    // erode: sum over 3x3 of dilated == 9
    float c9 = 0.f;
#pragma unroll
    for (int dy = 0; dy < 3; ++dy)
#pragma unroll
      for (int dx = 0; dx < 3; ++dx) c9 += sD[(ty + dy) * 34 + (tx + dx)];
    float er = (c9 > 8.5f) ? 1.0f : 0.0f;

    float x = sP[(ty + 1) * 34 + (tx + 1)];
    float t = sT[(ty + 3) * 38 + (tx + 3)];
    float L = log1pf(expf(-fabsf(x)));
    float mx0 = fmaxf(x, 0.0f);
    a0 += mx0 - x * t + L;    // bce(pred, target)
    a1 += mx0 - x * er + L;   // bce(pred, target_edges)

#define SPv(dy, dx) sS[(ty + (dy)) * 34 + (tx + (dx))]
#define DMv(dy, dx) sM[(ty + (dy)) * 34 + (tx + (dx))]
    float gxp = (SPv(0, 2) - SPv(0, 0)) + 2.f * (SPv(1, 2) - SPv(1, 0)) +
                (SPv(2, 2) - SPv(2, 0));
    float gyp = (SPv(2, 0) - SPv(0, 0)) + 2.f * (SPv(2, 1) - SPv(0, 1)) +
                (SPv(2, 2) - SPv(0, 2));
    float gp = sqrtf(gxp * gxp + gyp * gyp + EPSF);
    float gxd = (DMv(0, 2) - DMv(0, 0)) + 2.f * (DMv(1, 2) - DMv(1, 0)) +
                (DMv(2, 2) - DMv(2, 0));
    float gyd = (DMv(2, 0) - DMv(0, 0)) + 2.f * (DMv(2, 1) - DMv(0, 1)) +
                (DMv(2, 2) - DMv(0, 2));
    float gd = sqrtf(gxd * gxd + gyd * gyd + EPSF);
    float dc = DMv(1, 1);
    float lap = DMv(0, 1) + DMv(1, 0) + DMv(1, 2) + DMv(2, 1) - 4.f * dc;
    float cs = 1.0f / (1.0f + expf(-10.f * tanhf(0.1f * lap)));
    float sp = SPv(1, 1);
#undef SPv
#undef DMv
    a2 += gp; a3 += gp * gp; a4 += gd; a5 += gd * gd; a6 += gp * gd;
    a7 += sp * cs;
    a8 += dc; a9 += dc * dc;
    mnp = fminf(mnp, gp); mxp = fmaxf(mxp, gp);
    mnd = fminf(mnd, gd); mxd = fmaxf(mxd, gd);
  }

  // wave32 reduce, then cross-wave via LDS, then f64 atomics
  float v[14] = {a0, a1, a2, a3, a4, a5, a6, a7, a8, a9, mnp, mxp, mnd, mxd};
#pragma unroll
  for (int off = 16; off > 0; off >>= 1) {
#pragma unroll
    for (int k = 0; k < 10; ++k) v[k] += __shfl_down(v[k], off, 32);
    v[10] = fminf(v[10], __shfl_down(v[10], off, 32));
    v[11] = fmaxf(v[11], __shfl_down(v[11], off, 32));
    v[12] = fminf(v[12], __shfl_down(v[12], off, 32));
    v[13] = fmaxf(v[13], __shfl_down(v[13], off, 32));
  }
  int lane = threadIdx.x & 31, wid = threadIdx.x >> 5;
  if (lane == 0)
    for (int k = 0; k < 14; ++k) red[wid * 14 + k] = v[k];
  __syncthreads();
  if (threadIdx.x == 0) {
    float f[14];
    for (int k = 0; k < 14; ++k) f[k] = red[k];
    for (int w = 1; w < 8; ++w) {
      for (int k = 0; k < 10; ++k) f[k] += red[w * 14 + k];
      f[10] = fminf(f[10], red[w * 14 + 10]);
      f[11] = fmaxf(f[11], red[w * 14 + 11]);
      f[12] = fminf(f[12], red[w * 14 + 12]);
      f[13] = fmaxf(f[13], red[w * 14 + 13]);
    }
    for (int k = 0; k < 8; ++k) atomicAdd(&dws[k], (double)f[k]);
    atomicAdd(&dws[8 + b], (double)f[8]);
    atomicAdd(&dws[24 + b], (double)f[9]);
    atomicMin(&uws[0], __float_as_uint(f[10]));
    atomicMax(&uws[1], __float_as_uint(f[11]));
    atomicMin(&uws[2], __float_as_uint(f[12]));
    atomicMax(&uws[3], __float_as_uint(f[13]));
  }
}

__global__ __launch_bounds__(256)
void pass2_k(const float4* __restrict__ pred, const float4* __restrict__ dem,
             const double* __restrict__ dws, double* __restrict__ dsh) {
  __shared__ float red[8];
  const int b = blockIdx.z;
  const double n = 1048576.0;
  double Sd = dws[8 + b], Sd2 = dws[24 + b];
  double var = (Sd2 - Sd * Sd / n) / (n - 1.0);
  float mean = (float)(Sd / n);
  float inv = 1.0f / ((float)sqrt(var > 0.0 ? var : 0.0) + EPSF);
  size_t base4 = (size_t)b * (HW * HW / 4);
  size_t tg = (size_t)blockIdx.x * 256 + threadIdx.x;
  float acc = 0.f;
#pragma unroll
  for (int j = 0; j < 2; ++j) {
    size_t idx = base4 + tg * 2 + j;
    float4 p = pred[idx];
    float4 d = dem[idx];
    float z, s;
    z = (d.x - mean) * inv; s = 1.f / (1.f + expf(-p.x)); acc += s * expf(-0.5f * z * z);
    z = (d.y - mean) * inv; s = 1.f / (1.f + expf(-p.y)); acc += s * expf(-0.5f * z * z);
    z = (d.z - mean) * inv; s = 1.f / (1.f + expf(-p.z)); acc += s * expf(-0.5f * z * z);
    z = (d.w - mean) * inv; s = 1.f / (1.f + expf(-p.w)); acc += s * expf(-0.5f * z * z);
  }
#pragma unroll
  for (int off = 16; off > 0; off >>= 1) acc += __shfl_down(acc, off, 32);
  int lane = threadIdx.x & 31, wid = threadIdx.x >> 5;
  if (lane == 0) red[wid] = acc;
  __syncthreads();
  if (threadIdx.x == 0) {
    float f = 0.f;
    for (int w = 0; w < 8; ++w) f += red[w];
    atomicAdd(dsh, (double)f);
  }
}

__global__ void pass3_k(const double* dws, const unsigned* uws, float* out) {
  const double N = 16.0 * 1048576.0;
  double bce1 = dws[0] / N, bce2 = dws[1] / N;
  double m1p = dws[2] / N, m2p = dws[3] / N;
  double m1d = dws[4] / N, m2d = dws[5] / N, mpd = dws[6] / N;
  double curv = dws[7] / N, sh = dws[40] / N;
  float gminp = __uint_as_float(uws[0]), gmaxp = __uint_as_float(uws[1]);
  float gmind = __uint_as_float(uws[2]), gmaxd = __uint_as_float(uws[3]);
  double ap, bp, ad, bd;
  if (gmaxp > gminp) { ap = 1.0 / ((double)(gmaxp - gminp) + 1e-8); bp = -(double)gminp * ap; }
  else               { ap = 1.0; bp = 0.0; }
  if (gmaxd > gmind) { ad = 1.0 / ((double)(gmaxd - gmind) + 1e-8); bd = -(double)gmind * ad; }
  else               { ad = 1.0; bd = 0.0; }
  double c = bp - bd;
  double grad_cons = ap * ap * m2p + ad * ad * m2d - 2.0 * ap * ad * mpd +
                     2.0 * c * (ap * m1p - ad * m1d) + c * c;
  double geo = grad_cons + 0.5 * (-sh) + 0.3 * (-curv);
  out[0] = (float)(0.8 * bce1 + 0.1 * bce2 + 0.1 * geo);
}

extern "C" void kernel_launch(void* const* d_in, const int* in_sizes, int n_in,
                              void* d_out, int out_size, void* d_ws,
                              size_t ws_size, hipStream_t stream) {
  const float* pred = (const float*)d_in[0];
  const float* target = (const float*)d_in[1];
  const float* dem = (const float*)d_in[2];
  double* dws = (double*)d_ws;
  unsigned* uws = (unsigned*)((char*)d_ws + 41 * sizeof(double));

  init_ws_k<<<1, 64, 0, stream>>>(dws, uws);

  dim3 g1(HW / 32, HW / 32, NB);
  pass1_k<<<g1, 256, 0, stream>>>(pred, target, dem, dws, uws);

  dim3 g2(HW * HW / (256 * 8), 1, NB);
  pass2_k<<<g2, 256, 0, stream>>>((const float4*)pred, (const float4*)dem,
                                  dws, &dws[40]);

  pass3_k<<<1, 1, 0, stream>>>(dws, uws, (float*)d_out);
}